// GPR_EBM_32126355374957
// MI455X (gfx1250) — compile-verified
//
#include <hip/hip_runtime.h>
#include <hip/hip_bf16.h>

// ---------------------------------------------------------------------------
// GPR-GNN energy model, CDNA5 (gfx1250, wave32).
//   x1 = X @ W_in + b_in                        (WMMA f32 GEMM, K=256)
//   energy = (x1 @ We0 + be0) * temp0           (wave-per-node matvec)
//   for i in 0..2:
//     h  = x @ Wc[i] + bc[i]                    (WMMA f32 GEMM, K=128)
//     x  = leaky_relu( segment_sum(h[src]*w) )  (CSR wave-per-node aggregate)
//     energy += (x @ We[i+1] + be[i+1])*temp    (fused into aggregate epilogue)
// CSR is rebuilt every call (counting sort: histogram + LDS scan + scatter).
// ---------------------------------------------------------------------------

typedef float v2f __attribute__((ext_vector_type(2)));
typedef float v8f __attribute__((ext_vector_type(8)));

#define HIDC 128
#define SCAN_BS 1024

// ---------------- WMMA GEMM: C[M x 128] = A[M x K] @ B[K x 128] + bias -----
// One wave per 16x64 tile: blockDim = 64 (2 waves), block b -> rows b*16..+15,
// wave w -> cols w*64..+63.  V_WMMA_F32_16X16X4_F32, K stepped by 4.
__global__ __launch_bounds__(64)
void gemm_bias_wmma(const float* __restrict__ A, const float* __restrict__ B,
                    const float* __restrict__ bias, float* __restrict__ C,
                    int M, int K) {
  const int lane = threadIdx.x & 31;
  const int wave = threadIdx.x >> 5;          // 0..1
  const int r0   = blockIdx.x * 16;
  const int c0   = wave * 64;
  const int lr   = lane & 15;
  const bool hi  = lane >= 16;                // lanes 16..31 carry K+2/K+3 (A) / K rows 2,3 (B)
  if (r0 >= M) return;                        // wave-uniform

  v8f acc0 = {}, acc1 = {}, acc2 = {}, acc3 = {};
  const float* arow = A + (size_t)(r0 + lr) * K + (hi ? 2 : 0);

  for (int k = 0; k < K; k += 4) {
    v2f a;
    a.x = arow[k];
    a.y = arow[k + 1];
    const int kb = k + (hi ? 2 : 0);
    const float* brow0 = B + (size_t)kb * HIDC + c0 + lr;
    const float* brow1 = brow0 + HIDC;
    v2f b0; b0.x = brow0[0];  b0.y = brow1[0];
    v2f b1; b1.x = brow0[16]; b1.y = brow1[16];
    v2f b2; b2.x = brow0[32]; b2.y = brow1[32];
    v2f b3; b3.x = brow0[48]; b3.y = brow1[48];
    acc0 = __builtin_amdgcn_wmma_f32_16x16x4_f32(false, a, false, b0, (short)0, acc0, false, false);
    acc1 = __builtin_amdgcn_wmma_f32_16x16x4_f32(false, a, false, b1, (short)0, acc1, false, false);
    acc2 = __builtin_amdgcn_wmma_f32_16x16x4_f32(false, a, false, b2, (short)0, acc2, false, false);
    acc3 = __builtin_amdgcn_wmma_f32_16x16x4_f32(false, a, false, b3, (short)0, acc3, false, false);
  }

  // D layout: VGPR j -> row r0 + j + (hi?8:0), col = c0 + t*16 + lr
  v8f accs[4] = {acc0, acc1, acc2, acc3};
  #pragma unroll
  for (int t = 0; t < 4; ++t) {
    const int col = c0 + t * 16 + lr;
    const float bv = bias[col];
    #pragma unroll
    for (int j = 0; j < 8; ++j) {
      const int row = r0 + j + (hi ? 8 : 0);
      if (row < M) C[(size_t)row * HIDC + col] = accs[t][j] + bv;
    }
  }
}

// ---------------- CSR build --------------------------------------------------
__global__ void k_hist(const int* __restrict__ dst, int* __restrict__ deg, int E) {
  int i = blockIdx.x * blockDim.x + threadIdx.x;
  if (i < E) atomicAdd(&deg[dst[i]], 1);
}

__global__ __launch_bounds__(SCAN_BS)
void k_scan_local(const int* __restrict__ deg, int* __restrict__ offs,
                  int* __restrict__ aux, int N) {
  __shared__ int s[SCAN_BS];
  const int tid = threadIdx.x;
  const int i = blockIdx.x * SCAN_BS + tid;
  int v = (i < N) ? deg[i] : 0;
  s[tid] = v;
  __syncthreads();
  for (int o = 1; o < SCAN_BS; o <<= 1) {
    int t = (tid >= o) ? s[tid - o] : 0;
    __syncthreads();
    s[tid] += t;
    __syncthreads();
  }
  if (i < N) offs[i] = s[tid] - v;            // exclusive
  if (tid == SCAN_BS - 1) aux[blockIdx.x] = s[tid];
}

__global__ __launch_bounds__(128)
void k_scan_aux(const int* __restrict__ aux, int* __restrict__ auxo, int nb) {
  // nb <= 128 (N=100000 -> 98 blocks)
  __shared__ int s[128];
  const int tid = threadIdx.x;
  int v = (tid < nb) ? aux[tid] : 0;
  s[tid] = v;
  __syncthreads();
  for (int o = 1; o < 128; o <<= 1) {
    int t = (tid >= o) ? s[tid - o] : 0;
    __syncthreads();
    s[tid] += t;
    __syncthreads();
  }
  if (tid < nb) auxo[tid] = s[tid] - v;       // exclusive
}

__global__ void k_scan_add(int* __restrict__ offs, const int* __restrict__ auxo,
                           int* __restrict__ cursor, int N) {
  int i = blockIdx.x * blockDim.x + threadIdx.x;
  if (i < N) {
    int o = offs[i] + auxo[i / SCAN_BS];
    offs[i] = o;
    cursor[i] = o;
  }
}

__global__ void k_scatter(const int* __restrict__ src, const int* __restrict__ dst,
                          const float* __restrict__ w, int* __restrict__ cursor,
                          int* __restrict__ ssrc, float* __restrict__ sw, int E) {
  int i = blockIdx.x * blockDim.x + threadIdx.x;
  if (i < E) {
    int p = atomicAdd(&cursor[dst[i]], 1);
    ssrc[p] = src[i];
    sw[p] = w[i];
  }
}

// ---------------- energy term 0 (wave per node) -----------------------------
__global__ __launch_bounds__(256)
void k_energy0(const float* __restrict__ x, const float* __restrict__ We_l,
               const float* __restrict__ be, const float* __restrict__ temp,
               int l, float* __restrict__ energy, int N) {
  const int wid  = (blockIdx.x * blockDim.x + threadIdx.x) >> 5;
  const int lane = threadIdx.x & 31;
  if (wid >= N) return;
  const float4 xv = *(const float4*)(x + (size_t)wid * HIDC + lane * 4);
  const float4 wv = *(const float4*)(We_l + lane * 4);
  float p = xv.x * wv.x + xv.y * wv.y + xv.z * wv.z + xv.w * wv.w;
  #pragma unroll
  for (int off = 16; off > 0; off >>= 1) p += __shfl_xor(p, off, 32);
  if (lane == 0) energy[wid] = (p + be[l]) * temp[l];
}

// ------- aggregation: x[n] = leaky(sum_e w*h[src]); fused energy term -------
__global__ __launch_bounds__(256)
void k_aggregate(const float* __restrict__ h, const int* __restrict__ ssrc,
                 const float* __restrict__ sw, const int* __restrict__ offs,
                 const int* __restrict__ deg, float* __restrict__ xout,
                 const float* __restrict__ We_l, const float* __restrict__ be,
                 const float* __restrict__ temp, int l,
                 float* __restrict__ energy, int N) {
  const int wid  = (blockIdx.x * blockDim.x + threadIdx.x) >> 5;
  const int lane = threadIdx.x & 31;
  if (wid >= N) return;
  const int beg = offs[wid];
  const int cnt = deg[wid];
  float4 acc = make_float4(0.f, 0.f, 0.f, 0.f);
  for (int e = 0; e < cnt; ++e) {
    const int   s = ssrc[beg + e];               // wave-uniform -> broadcast
    const float w = sw[beg + e];
    const float4 hv = *(const float4*)(h + (size_t)s * HIDC + lane * 4);
    acc.x += w * hv.x;
    acc.y += w * hv.y;
    acc.z += w * hv.z;
    acc.w += w * hv.w;
  }
  acc.x = acc.x > 0.f ? acc.x : 0.01f * acc.x;   // leaky_relu(0.01)
  acc.y = acc.y > 0.f ? acc.y : 0.01f * acc.y;
  acc.z = acc.z > 0.f ? acc.z : 0.01f * acc.z;
  acc.w = acc.w > 0.f ? acc.w : 0.01f * acc.w;
  *(float4*)(xout + (size_t)wid * HIDC + lane * 4) = acc;

  const float4 wv = *(const float4*)(We_l + lane * 4);
  float p = acc.x * wv.x + acc.y * wv.y + acc.z * wv.z + acc.w * wv.w;
  #pragma unroll
  for (int off = 16; off > 0; off >>= 1) p += __shfl_xor(p, off, 32);
  if (lane == 0) energy[wid] += (p + be[l]) * temp[l];
}

// ---------------------------------------------------------------------------
extern "C" void kernel_launch(void* const* d_in, const int* in_sizes, int n_in,
                              void* d_out, int out_size, void* d_ws, size_t ws_size,
                              hipStream_t stream) {
  const float* x_in = (const float*)d_in[0];   // [N, 256]
  const int*   esrc = (const int*)  d_in[1];   // [E]
  const int*   edst = (const int*)  d_in[2];   // [E]
  const float* ew   = (const float*)d_in[3];   // [E]
  const float* W_in = (const float*)d_in[4];   // [256, 128]
  const float* b_in = (const float*)d_in[5];   // [128]
  const float* Wc   = (const float*)d_in[6];   // [3, 128, 128]
  const float* bc   = (const float*)d_in[7];   // [3, 128]
  const float* We   = (const float*)d_in[8];   // [4, 128]
  const float* be   = (const float*)d_in[9];   // [4]
  const float* temp = (const float*)d_in[10];  // [4]
  float* energy = (float*)d_out;               // [N]

  const int INC = 256;
  const int N = in_sizes[0] / INC;             // 100000
  const int E = in_sizes[1];                   // 1600000
  const int L = 3;

  // ---- workspace layout (all 16B-aligned chunks) ----
  char* w = (char*)d_ws;
  float* xbuf = (float*)w;                      w += (size_t)N * HIDC * 4;
  float* hbuf = (float*)w;                      w += (size_t)N * HIDC * 4;
  float* sw   = (float*)w;                      w += (size_t)E * 4;
  int*   ssrc = (int*)w;                        w += (size_t)E * 4;
  int*   deg  = (int*)w;                        w += (size_t)N * 4;
  int*   offs = (int*)w;                        w += (size_t)N * 4;
  int*   curs = (int*)w;                        w += (size_t)N * 4;
  int*   aux  = (int*)w;                        w += 512;
  int*   auxo = (int*)w;                        w += 512;

  // ---- 1) CSR build (counting sort by edge_dst) ----
  hipMemsetAsync(deg, 0, (size_t)N * 4, stream);
  k_hist<<<(E + 255) / 256, 256, 0, stream>>>(edst, deg, E);
  const int nb = (N + SCAN_BS - 1) / SCAN_BS;   // 98 (<=128)
  k_scan_local<<<nb, SCAN_BS, 0, stream>>>(deg, offs, aux, N);
  k_scan_aux<<<1, 128, 0, stream>>>(aux, auxo, nb);
  k_scan_add<<<(N + 255) / 256, 256, 0, stream>>>(offs, auxo, curs, N);
  k_scatter<<<(E + 255) / 256, 256, 0, stream>>>(esrc, edst, ew, curs, ssrc, sw, E);

  // ---- 2) input linear: xbuf = X @ W_in + b_in ----
  gemm_bias_wmma<<<(N + 15) / 16, 64, 0, stream>>>(x_in, W_in, b_in, xbuf, N, INC);

  // ---- 3) energy term 0 ----
  const int wpg = ((N * 32) + 255) / 256;       // wave-per-node grids
  k_energy0<<<wpg, 256, 0, stream>>>(xbuf, We + 0, be, temp, 0, energy, N);

  // ---- 4) layers ----
  for (int i = 0; i < L; ++i) {
    gemm_bias_wmma<<<(N + 15) / 16, 64, 0, stream>>>(
        xbuf, Wc + (size_t)i * HIDC * HIDC, bc + (size_t)i * HIDC, hbuf, N, HIDC);
    k_aggregate<<<wpg, 256, 0, stream>>>(
        hbuf, ssrc, sw, offs, deg, xbuf,
        We + (size_t)(i + 1) * HIDC, be, temp, i + 1, energy, N);
  }
}